// MonarchMatrix_73615739453579
// MI455X (gfx1250) — compile-verified
//
#include <hip/hip_runtime.h>

typedef __attribute__((ext_vector_type(2))) float v2f;
typedef __attribute__((ext_vector_type(8))) float v8f;

#define SIZE 1024
#define NBLK 16
#define BS 64
#define ROWS 32768           // BATCH*SEQ = 8*4096
#define WG_ROWS 128
#define LDS_STRIDE 68        // pad so b64 row-gather is bank-conflict free (64 banks)

__device__ __forceinline__ v8f wmma_f32_4(v2f a, v2f b, v8f c) {
  // 8 args: (neg_a, A, neg_b, B, c_mod, C, reuse_a, reuse_b)
  return __builtin_amdgcn_wmma_f32_16x16x4_f32(false, a, false, b, (short)0, c,
                                               false, false);
}

// ---------------------------------------------------------------------------
// Kernel 1: collapse P (10 layers of adjacent-pair 2x2), B, Q into a single
// block-diagonal matrix M_n = Peff_n * B_n * Qeff_n, and store it already
// swizzled into WMMA B-fragment layout:
//   frag[n][ (ks*4 + ntile)*32 + lane ] = { M[4ks + 2*(lane>=16)][ntile*16 + lane%16],
//                                           M[4ks + 2*(lane>=16) + 1][...] }
// Total: 16 blocks * 2048 float2 = 256 KB in workspace.
// ---------------------------------------------------------------------------
__global__ __launch_bounds__(64) void monarch_fuse_73615739453579(
    const float* __restrict__ Pw, const float* __restrict__ Qw,
    const float* __restrict__ Bb, float2* __restrict__ frag) {
  __shared__ float sPe[32][4];
  __shared__ float sQe[32][4];
  __shared__ float sB[64][64];
  __shared__ float sT[64][64];
  const int n = blockIdx.x;
  const int t = threadIdx.x;

  {  // threads 0..31: chain-multiply P 2x2s; threads 32..63: Q
    const float* W = (t < 32) ? Pw : Qw;
    const int pr = t & 31;
    const int p = n * 32 + pr;              // global pair index in [0,512)
    const float* w0 = W + (size_t)p * 4;
    float e00 = w0[0], e01 = w0[1], e10 = w0[2], e11 = w0[3];
#pragma unroll
    for (int l = 1; l < 10; ++l) {          // E <- E * W_l  (row-vector conv.)
      const float* wl = W + ((size_t)l * 512 + p) * 4;
      const float w00 = wl[0], w01 = wl[1], w10 = wl[2], w11 = wl[3];
      const float n00 = e00 * w00 + e01 * w10;
      const float n01 = e00 * w01 + e01 * w11;
      const float n10 = e10 * w00 + e11 * w10;
      const float n11 = e10 * w01 + e11 * w11;
      e00 = n00; e01 = n01; e10 = n10; e11 = n11;
    }
    float* dst = (t < 32) ? sPe[pr] : sQe[pr];
    dst[0] = e00; dst[1] = e01; dst[2] = e10; dst[3] = e11;
  }
  for (int i = t; i < 64 * 64; i += 64) sB[i >> 6][i & 63] = Bb[(size_t)n * 4096 + i];
  __syncthreads();

  // T = Peff * B  (thread t owns column t; Peff mixes row pairs)
#pragma unroll 4
  for (int k = 0; k < 32; ++k) {
    const float b0 = sB[2 * k][t], b1 = sB[2 * k + 1][t];
    sT[2 * k][t]     = sPe[k][0] * b0 + sPe[k][1] * b1;
    sT[2 * k + 1][t] = sPe[k][2] * b0 + sPe[k][3] * b1;
  }
  __syncthreads();

  // M = T * Qeff  (thread t owns row t; Qeff mixes column pairs) -> reuse sB
#pragma unroll 4
  for (int k = 0; k < 32; ++k) {
    const float t0 = sT[t][2 * k], t1 = sT[t][2 * k + 1];
    sB[t][2 * k]     = t0 * sQe[k][0] + t1 * sQe[k][2];
    sB[t][2 * k + 1] = t0 * sQe[k][1] + t1 * sQe[k][3];
  }
  __syncthreads();

  // Emit WMMA B-fragment layout (2048 float2 per block)
  for (int i = t; i < 16 * 4 * 32; i += 64) {
    const int lane = i & 31;
    const int nt = (i >> 5) & 3;
    const int ks = i >> 7;
    const int r0 = 4 * ks + 2 * (lane >> 4);
    const int c = nt * 16 + (lane & 15);
    frag[(size_t)n * 2048 + i] = make_float2(sB[r0][c], sB[r0 + 1][c]);
  }
}

// ---------------------------------------------------------------------------
// Kernel 2: out[:, 64n:64n+64] = x[:, 64n:64n+64] @ M_n via f32 WMMA.
// Grid (256, 16), 256 threads (8 wave32). Each WG stages 128 rows x 64 cols of
// x into LDS (each x element read exactly once device-wide), each wave does a
// 16x64 tile: 16 K-steps x 4 N-tiles of v_wmma_f32_16x16x4_f32.
// ---------------------------------------------------------------------------
__global__ __launch_bounds__(256) void monarch_main_73615739453579(
    const float* __restrict__ x, const float2* __restrict__ frag,
    float* __restrict__ out) {
  __shared__ float sA[WG_ROWS * LDS_STRIDE];
  const int blk = blockIdx.y;
  const int rowbase = blockIdx.x * WG_ROWS;
  const int t = threadIdx.x;

  // Cooperative coalesced load: 128 rows x 16 float4
  for (int i = t; i < WG_ROWS * 16; i += 256) {
    const int r = i >> 4, c4 = i & 15;
    const float4 v = *(const float4*)(x + (size_t)(rowbase + r) * SIZE +
                                      blk * BS + c4 * 4);
    *(float4*)(sA + r * LDS_STRIDE + c4 * 4) = v;
  }
  __syncthreads();

  const int wave = t >> 5;
  const int lane = t & 31;
  const int half = lane >> 4;   // 0: K={0,1} / M rows 0-7 of C; 1: K={2,3} / rows 8-15
  const int ln = lane & 15;
  const float* aBase = sA + (wave * 16 + ln) * LDS_STRIDE + 2 * half;
  const float2* bf = frag + (size_t)blk * 2048 + lane;

  v8f acc0 = {}, acc1 = {}, acc2 = {}, acc3 = {};
#pragma unroll
  for (int ks = 0; ks < 16; ++ks) {
    const float2 af = *(const float2*)(aBase + ks * 4);
    v2f a; a.x = af.x; a.y = af.y;
    float2 b; v2f bv;
    b = bf[(ks * 4 + 0) * 32]; bv.x = b.x; bv.y = b.y; acc0 = wmma_f32_4(a, bv, acc0);
    b = bf[(ks * 4 + 1) * 32]; bv.x = b.x; bv.y = b.y; acc1 = wmma_f32_4(a, bv, acc1);
    b = bf[(ks * 4 + 2) * 32]; bv.x = b.x; bv.y = b.y; acc2 = wmma_f32_4(a, bv, acc2);
    b = bf[(ks * 4 + 3) * 32]; bv.x = b.x; bv.y = b.y; acc3 = wmma_f32_4(a, bv, acc3);
  }

  // C/D layout: VGPR v holds M = v + 8*half, N = ln (per 16-wide N tile)
  const int gr0 = rowbase + wave * 16 + half * 8;
  float* obase = out + (size_t)gr0 * SIZE + blk * BS + ln;
#pragma unroll
  for (int v = 0; v < 8; ++v) {
    float* orow = obase + (size_t)v * SIZE;
    orow[0]  = acc0[v];
    orow[16] = acc1[v];
    orow[32] = acc2[v];
    orow[48] = acc3[v];
  }
}

extern "C" void kernel_launch(void* const* d_in, const int* in_sizes, int n_in,
                              void* d_out, int out_size, void* d_ws, size_t ws_size,
                              hipStream_t stream) {
  (void)in_sizes; (void)n_in; (void)out_size; (void)ws_size;
  const float* x  = (const float*)d_in[0];
  const float* Pw = (const float*)d_in[1];
  const float* Qw = (const float*)d_in[2];
  const float* Bb = (const float*)d_in[3];
  float2* frag = (float2*)d_ws;  // 16 * 2048 * 8 B = 256 KB scratch

  monarch_fuse_73615739453579<<<dim3(NBLK), dim3(64), 0, stream>>>(Pw, Qw, Bb, frag);
  monarch_main_73615739453579<<<dim3(ROWS / WG_ROWS, NBLK), dim3(256), 0, stream>>>(
      x, frag, (float*)d_out);
}